// AMT_S_283467841956
// MI455X (gfx1250) — compile-verified
//
#include <hip/hip_runtime.h>
#include <hip/hip_bf16.h>
#include <math.h>

typedef __attribute__((ext_vector_type(2))) float v2f;
typedef __attribute__((ext_vector_type(8))) float v8f;

#define GHW 4096      // h*w = 64*64
#define GD  84        // feature dim
#define IMH 512
#define IMW 512
#define IMHW 262144

// ---------------------------------------------------------------------------
// Async Global->LDS copy (CDNA5 GLOBAL_LOAD_ASYNC_TO_LDS_B128, ASYNCcnt
// tracked). The builtin takes <4 x i32>* in AS1 (global) and AS3 (LDS).
// ---------------------------------------------------------------------------
#if __has_builtin(__builtin_amdgcn_global_load_async_to_lds_b128)
#define HAS_ASYNC_LDS 1
#else
#define HAS_ASYNC_LDS 0
#endif

#if HAS_ASYNC_LDS
typedef __attribute__((__vector_size__(4 * sizeof(int)))) int v4i_t;
typedef __attribute__((address_space(1))) v4i_t gas1_v4i;
typedef __attribute__((address_space(3))) v4i_t las3_v4i;

__device__ __forceinline__ void async_copy16(const float* g, float* l) {
    __builtin_amdgcn_global_load_async_to_lds_b128(
        (gas1_v4i*)(void*)g,
        (las3_v4i*)(void*)l,
        0, 0);
}
__device__ __forceinline__ void wait_async_zero() {
#if __has_builtin(__builtin_amdgcn_s_wait_asynccnt)
    __builtin_amdgcn_s_wait_asynccnt(0);
#else
    asm volatile("s_wait_asynccnt 0x0" ::: "memory");
#endif
}
#endif

// ---------------------------------------------------------------------------
// Kernel 1: all-pairs correlation via V_WMMA_F32_16X16X4_F32.
// corr[b][i][j] = sum_c f0[b][c][i] * f1[b][c][j] / sqrt(84)
// Each block: 64x64 output tile, 8 waves, each wave owns two 16x16 subtiles
// (same A fragment, two B fragments). corr_T written via an LDS-transposed
// restage so both outputs stream out as coalesced rows.
// ---------------------------------------------------------------------------
__global__ __launch_bounds__(256)
void corr_wmma_kernel(const float* __restrict__ f0, const float* __restrict__ f1,
                      float* __restrict__ corr, float* __restrict__ corrT) {
    const int LDP = 72;                      // padded row stride (288B, 16B-aligned)
    __shared__ __align__(16) float As[GD * LDP];
    __shared__ __align__(16) float Bs[GD * LDP];

    const int b  = blockIdx.z;
    const int i0 = blockIdx.y * 64;
    const int j0 = blockIdx.x * 64;
    const int tid = threadIdx.x;

    const float* A  = f0 + (size_t)b * GD * GHW;   // [k][i]
    const float* Bm = f1 + (size_t)b * GD * GHW;   // [k][j]

    // cooperative load of 84x64 panels (float4 granule, coalesced rows)
#if HAS_ASYNC_LDS
    for (int v = tid; v < GD * 16; v += 256) {
        int k  = v >> 4;
        int c4 = (v & 15) << 2;
        async_copy16(A  + (size_t)k * GHW + i0 + c4, &As[k * LDP + c4]);
        async_copy16(Bm + (size_t)k * GHW + j0 + c4, &Bs[k * LDP + c4]);
    }
    wait_async_zero();
#else
    for (int v = tid; v < GD * 16; v += 256) {
        int k  = v >> 4;
        int c4 = (v & 15) << 2;
        float4 av = *(const float4*)(A  + (size_t)k * GHW + i0 + c4);
        float4 bv = *(const float4*)(Bm + (size_t)k * GHW + j0 + c4);
        *(float4*)(&As[k * LDP + c4]) = av;
        *(float4*)(&Bs[k * LDP + c4]) = bv;
    }
#endif
    __syncthreads();

    const int wv    = tid >> 5;          // wave 0..7
    const int lane  = tid & 31;
    const int r     = wv & 3;            // row-subtile 0..3
    const int cpair = wv >> 2;           // col-pair 0..1
    const int mrow  = r * 16 + (lane & 15);
    const int koff  = (lane >> 4) * 2;   // lanes 16-31 hold K+2/K+3
    const int n0    = cpair * 32 + (lane & 15);
    const int n1    = n0 + 16;

    v8f acc0 = {};
    v8f acc1 = {};
    for (int k = 0; k < GD; k += 4) {
        v2f a, b0, b1;
        a.x  = As[(k + koff    ) * LDP + mrow];
        a.y  = As[(k + koff + 1) * LDP + mrow];
        b0.x = Bs[(k + koff    ) * LDP + n0];
        b0.y = Bs[(k + koff + 1) * LDP + n0];
        b1.x = Bs[(k + koff    ) * LDP + n1];
        b1.y = Bs[(k + koff + 1) * LDP + n1];
        acc0 = __builtin_amdgcn_wmma_f32_16x16x4_f32(false, a, false, b0, (short)0, acc0, false, false);
        acc1 = __builtin_amdgcn_wmma_f32_16x16x4_f32(false, a, false, b1, (short)0, acc1, false, false);
    }

    const float scale = 0.10910894511799618f;  // 1/sqrt(84)
    const size_t cb = (size_t)b * GHW * GHW;
    const int nlo = lane & 15;
    const int mhalf = (lane >> 4) * 8;          // C/D: lanes 16-31 hold M+8

    // corr: direct stores (16 consecutive floats per lane-half -> coalesced)
#pragma unroll
    for (int v = 0; v < 8; v++) {
        int m  = i0 + r * 16 + mhalf + v;
        int nA = j0 + cpair * 32 + nlo;
        corr[cb + (size_t)m * GHW + nA     ] = acc0[v] * scale;
        corr[cb + (size_t)m * GHW + nA + 16] = acc1[v] * scale;
    }

    // corr_T: restage transposed through LDS (reuse As), then coalesced rows.
    __syncthreads();                 // all waves done reading As/Bs
    float* tile = As;                // 64 x 68 (padded) fits in As (6048 floats)
    const int jl0 = cpair * 32 + nlo;
    const int il0 = r * 16 + mhalf;
#pragma unroll
    for (int v = 0; v < 8; v++) {
        tile[(jl0     ) * 68 + il0 + v] = acc0[v] * scale;
        tile[(jl0 + 16) * 68 + il0 + v] = acc1[v] * scale;
    }
    __syncthreads();
    for (int i = tid; i < 4096; i += 256) {
        int jl = i >> 6, il = i & 63;
        corrT[cb + (size_t)(j0 + jl) * GHW + i0 + il] = tile[jl * 68 + il];
    }
}

// ---------------------------------------------------------------------------
// Kernel 2: avg-pool pyramid (levels 1..3) per 64x64 correlation slice.
// One block per (direction, entry); level-0 pulled in via async LDS fill.
// ---------------------------------------------------------------------------
__global__ __launch_bounds__(256)
void pyramid_kernel(const float* __restrict__ corr, const float* __restrict__ corrT,
                    float* __restrict__ pyr, float* __restrict__ pyrT) {
    __shared__ __align__(16) float l0[4096];
    __shared__ float l1[1024];
    __shared__ float l2[256];

    const int e   = blockIdx.x;          // 0..16383
    const int dir = e >> 13;
    const int ent = e & 8191;
    const int tid = threadIdx.x;

    const float* src = (dir ? corrT : corr) + (size_t)ent * GHW;
    float* dst       = (dir ? pyrT  : pyr ) + (size_t)ent * 1344;

#if HAS_ASYNC_LDS
#pragma unroll
    for (int it = 0; it < 4; it++) {
        int i4 = it * 256 + tid;             // 1024 float4 granules
        async_copy16(src + i4 * 4, &l0[i4 * 4]);
    }
    wait_async_zero();
#else
    for (int i = tid; i < 4096; i += 256) l0[i] = src[i];
#endif
    __syncthreads();

    for (int i = tid; i < 1024; i += 256) {
        int y = i >> 5, x = i & 31;
        float v = 0.25f * (l0[(2*y)*64 + 2*x] + l0[(2*y)*64 + 2*x + 1] +
                           l0[(2*y+1)*64 + 2*x] + l0[(2*y+1)*64 + 2*x + 1]);
        l1[i] = v;
        dst[i] = v;
    }
    __syncthreads();

    if (tid < 256) {
        int y = tid >> 4, x = tid & 15;
        float v = 0.25f * (l1[(2*y)*32 + 2*x] + l1[(2*y)*32 + 2*x + 1] +
                           l1[(2*y+1)*32 + 2*x] + l1[(2*y+1)*32 + 2*x + 1]);
        l2[tid] = v;
        dst[1024 + tid] = v;
    }
    __syncthreads();

    if (tid < 64) {
        int y = tid >> 3, x = tid & 7;
        float v = 0.25f * (l2[(2*y)*16 + 2*x] + l2[(2*y)*16 + 2*x + 1] +
                           l2[(2*y+1)*16 + 2*x] + l2[(2*y+1)*16 + 2*x + 1]);
        dst[1280 + tid] = v;
    }
}

// ---------------------------------------------------------------------------
// Kernel 3: correlation lookup (bilinear, 'zeros' padding).
// One thread per output element; output index == d_out flat index for
// (corr_feat, corr_feat_T) in return order.
// ---------------------------------------------------------------------------
__global__ __launch_bounds__(256)
void corr_lookup_kernel(const float* __restrict__ corr, const float* __restrict__ corrT,
                        const float* __restrict__ pyr, const float* __restrict__ pyrT,
                        const float* __restrict__ flow0_lr, const float* __restrict__ flow1_lr,
                        float* __restrict__ out) {
    const int idx = blockIdx.x * 256 + threadIdx.x;
    const int PER_DIR = 2 * 324 * GHW;    // 2,654,208
    if (idx >= 2 * PER_DIR) return;

    const int dir = idx / PER_DIR;
    int r0 = idx - dir * PER_DIR;
    const int b = r0 / (324 * GHW);
    int r1 = r0 - b * (324 * GHW);
    const int ch = r1 >> 12;
    const int p  = r1 & 4095;
    const int l  = ch / 81;
    const int s  = ch - l * 81;
    const int a  = s / 9;                 // delta[...,0] -> added to x  (dy grid)
    const int c  = s - a * 9;             // delta[...,1] -> added to y  (dx grid)
    const int y = p >> 6, x = p & 63;

    const float* flow = (dir ? flow1_lr : flow0_lr) + (size_t)b * 2 * GHW;
    const float fx = flow[p];
    const float fy = flow[GHW + p];
    const float inv = 1.0f / (float)(1 << l);
    const float xs = ((float)x + fx) * inv + (float)(a - 4);
    const float ys = ((float)y + fy) * inv + (float)(c - 4);

    const int ent = b * GHW + p;
    const float* img;
    int dim;
    if (l == 0) {
        img = (dir ? corrT : corr) + (size_t)ent * GHW;
        dim = 64;
    } else {
        const int off = (l == 1) ? 0 : ((l == 2) ? 1024 : 1280);
        img = (dir ? pyrT : pyr) + (size_t)ent * 1344 + off;
        dim = 64 >> l;
    }

    const float x0f = floorf(xs), y0f = floorf(ys);
    const float wx = xs - x0f, wy = ys - y0f;
    const int x0 = (int)x0f, y0 = (int)y0f;
    float acc = 0.0f;
#pragma unroll
    for (int cy = 0; cy < 2; cy++) {
#pragma unroll
        for (int cx = 0; cx < 2; cx++) {
            const int xi = x0 + cx, yi = y0 + cy;
            const float wgt = (cx ? wx : 1.0f - wx) * (cy ? wy : 1.0f - wy);
            if (xi >= 0 && xi < dim && yi >= 0 && yi < dim)
                acc += img[yi * dim + xi] * wgt;
        }
    }
    out[idx] = acc;
}

// ---------------------------------------------------------------------------
// Kernel 4: per-flow warp (border bilinear) + mask blend + mean + residual.
// One thread per (b*n, y, x), computing 3 channels.
// ---------------------------------------------------------------------------
__device__ __forceinline__
void bil3_border(const float* base, float xs, float ys, float o[3]) {
    xs = fminf(fmaxf(xs, 0.0f), (float)(IMW - 1));
    ys = fminf(fmaxf(ys, 0.0f), (float)(IMH - 1));
    const float x0f = floorf(xs), y0f = floorf(ys);
    const float wx = xs - x0f, wy = ys - y0f;
    const int x0 = (int)x0f, y0 = (int)y0f;
    const int x1 = min(x0 + 1, IMW - 1), y1 = min(y0 + 1, IMH - 1);
    const float w00 = (1.0f - wx) * (1.0f - wy), w10 = wx * (1.0f - wy);
    const float w01 = (1.0f - wx) * wy,          w11 = wx * wy;
    const int i00 = y0 * IMW + x0, i10 = y0 * IMW + x1;
    const int i01 = y1 * IMW + x0, i11 = y1 * IMW + x1;
#pragma unroll
    for (int ch = 0; ch < 3; ch++) {
        const float* pl = base + (size_t)ch * IMHW;
        o[ch] = pl[i00] * w00 + pl[i10] * w10 + pl[i01] * w01 + pl[i11] * w11;
    }
}

__global__ __launch_bounds__(256)
void warp_combine_kernel(const float* __restrict__ img0, const float* __restrict__ img1,
                         const float* __restrict__ flow0, const float* __restrict__ flow1,
                         const float* __restrict__ mask, const float* __restrict__ img_res,
                         const float* __restrict__ mean, float* __restrict__ warps) {
    const int idx = blockIdx.x * 256 + threadIdx.x;   // 6 * 262144 threads
    const int bn = idx >> 18;
    const int p  = idx & (IMHW - 1);
    const int b = bn / 3, f = bn - b * 3;
    const int y = p >> 9, x = p & 511;

    const float* F0 = flow0 + ((size_t)b * 6 + 2 * f) * IMHW;
    const float* F1 = flow1 + ((size_t)b * 6 + 2 * f) * IMHW;
    const float f0x = F0[p], f0y = F0[IMHW + p];
    const float f1x = F1[p], f1y = F1[IMHW + p];
    const float m  = mask[((size_t)b * 3 + f) * IMHW + p];
    const float mn = mean[b];

    float s0[3], s1[3];
    bil3_border(img0 + (size_t)b * 3 * IMHW, (float)x + f0x, (float)y + f0y, s0);
    bil3_border(img1 + (size_t)b * 3 * IMHW, (float)x + f1x, (float)y + f1y, s1);

#pragma unroll
    for (int ch = 0; ch < 3; ch++) {
        const float res = img_res[((size_t)b * 9 + f * 3 + ch) * IMHW + p];
        warps[((size_t)bn * 3 + ch) * IMHW + p] = m * s0[ch] + (1.0f - m) * s1[ch] + mn + res;
    }
}

// ---------------------------------------------------------------------------
// Kernel 5: 7x7 conv 9->18 + PReLU. 16x16 output tile per block; halo patch
// and all weights in LDS; each thread accumulates all 18 output channels.
// ---------------------------------------------------------------------------
__global__ __launch_bounds__(256)
void conv1_kernel(const float* __restrict__ warps, const float* __restrict__ w1,
                  const float* __restrict__ b1, const float* __restrict__ a1,
                  float* __restrict__ comb) {
    __shared__ float patch[9 * 22 * 22];
    __shared__ float wt[18 * 441];
    __shared__ float bias[18];
    __shared__ float alpha[18];

    const int b  = blockIdx.z;
    const int bx = blockIdx.x * 16, by = blockIdx.y * 16;
    const int tid = threadIdx.y * 16 + threadIdx.x;

    for (int i = tid; i < 18 * 441; i += 256) wt[i] = w1[i];
    if (tid < 18) { bias[tid] = b1[tid]; alpha[tid] = a1[tid]; }

    const float* inb = warps + (size_t)b * 9 * IMHW;
    for (int i = tid; i < 9 * 484; i += 256) {
        const int ic = i / 484;
        const int pp = i - ic * 484;
        const int py = pp / 22, px = pp - py * 22;
        const int gy = by + py - 3, gx = bx + px - 3;
        float v = 0.0f;
        if (gy >= 0 && gy < IMH && gx >= 0 && gx < IMW)
            v = inb[(size_t)ic * IMHW + gy * IMW + gx];
        patch[i] = v;
    }
    __syncthreads();

    float acc[18];
#pragma unroll
    for (int oc = 0; oc < 18; oc++) acc[oc] = 0.0f;

    const int ty = threadIdx.y, tx = threadIdx.x;
    for (int ic = 0; ic < 9; ic++) {
        for (int ky = 0; ky < 7; ky++) {
#pragma unroll
            for (int kx = 0; kx < 7; kx++) {
                const float pv = patch[ic * 484 + (ty + ky) * 22 + (tx + kx)];
                const float* wr = &wt[ic * 49 + ky * 7 + kx];
#pragma unroll
                for (int oc = 0; oc < 18; oc++)
                    acc[oc] += pv * wr[oc * 441];
            }
        }
    }

    const int y = by + ty, x = bx + tx;
#pragma unroll
    for (int oc = 0; oc < 18; oc++) {
        float v = acc[oc] + bias[oc];
        v = (v >= 0.0f) ? v : alpha[oc] * v;
        comb[((size_t)(b * 18 + oc)) * IMHW + y * IMW + x] = v;
    }
}

// ---------------------------------------------------------------------------
// Kernel 6: 7x7 conv 18->3 + warps.mean(axis=1) -> imgt_pred.
// ---------------------------------------------------------------------------
__global__ __launch_bounds__(256)
void conv2_kernel(const float* __restrict__ comb, const float* __restrict__ w2,
                  const float* __restrict__ b2, const float* __restrict__ warps,
                  float* __restrict__ imgt) {
    __shared__ float patch[18 * 484];
    __shared__ float wt[3 * 18 * 49];
    __shared__ float bias[3];

    const int b  = blockIdx.z;
    const int bx = blockIdx.x * 16, by = blockIdx.y * 16;
    const int tid = threadIdx.y * 16 + threadIdx.x;

    for (int i = tid; i < 3 * 18 * 49; i += 256) wt[i] = w2[i];
    if (tid < 3) bias[tid] = b2[tid];

    const float* inb = comb + (size_t)b * 18 * IMHW;
    for (int i = tid; i < 18 * 484; i += 256) {
        const int ic = i / 484;
        const int pp = i - ic * 484;
        const int py = pp / 22, px = pp - py * 22;
        const int gy = by + py - 3, gx = bx + px - 3;
        float v = 0.0f;
        if (gy >= 0 && gy < IMH && gx >= 0 && gx < IMW)
            v = inb[(size_t)ic * IMHW + gy * IMW + gx];
        patch[i] = v;
    }
    __syncthreads();

    float acc[3] = {0.0f, 0.0f, 0.0f};
    const int ty = threadIdx.y, tx = threadIdx.x;
    for (int ic = 0; ic < 18; ic++) {
        for (int ky = 0; ky < 7; ky++) {
#pragma unroll
            for (int kx = 0; kx < 7; kx++) {
                const float pv = patch[ic * 484 + (ty + ky) * 22 + (tx + kx)];
                const float* wr = &wt[ic * 49 + ky * 7 + kx];
#pragma unroll
                for (int oc = 0; oc < 3; oc++)
                    acc[oc] += pv * wr[oc * 18 * 49];
            }
        }
    }

    const int y = by + ty, x = bx + tx;
    const int p = y * IMW + x;
#pragma unroll
    for (int oc = 0; oc < 3; oc++) {
        float mw = (warps[((size_t)(b * 3 + 0) * 3 + oc) * IMHW + p] +
                    warps[((size_t)(b * 3 + 1) * 3 + oc) * IMHW + p] +
                    warps[((size_t)(b * 3 + 2) * 3 + oc) * IMHW + p]) * (1.0f / 3.0f);
        imgt[((size_t)(b * 3 + oc)) * IMHW + p] = mw + acc[oc] + bias[oc];
    }
}

// ---------------------------------------------------------------------------
// Launch
// ---------------------------------------------------------------------------
extern "C" void kernel_launch(void* const* d_in, const int* in_sizes, int n_in,
                              void* d_out, int out_size, void* d_ws, size_t ws_size,
                              hipStream_t stream) {
    const float* img0     = (const float*)d_in[0];
    const float* img1     = (const float*)d_in[1];
    const float* fmap0    = (const float*)d_in[2];
    const float* fmap1    = (const float*)d_in[3];
    const float* flow0_lr = (const float*)d_in[4];
    const float* flow1_lr = (const float*)d_in[5];
    const float* flow0    = (const float*)d_in[6];
    const float* flow1    = (const float*)d_in[7];
    const float* mask     = (const float*)d_in[8];
    const float* img_res  = (const float*)d_in[9];
    const float* mean     = (const float*)d_in[10];
    const float* w1       = (const float*)d_in[11];
    const float* b1       = (const float*)d_in[12];
    const float* a1       = (const float*)d_in[13];
    const float* w2       = (const float*)d_in[14];
    const float* b2       = (const float*)d_in[15];

    float* ws    = (float*)d_ws;
    float* corr  = ws;                      // 33,554,432 floats
    float* corrT = ws + 33554432ull;        // 33,554,432
    float* pyr   = ws + 67108864ull;        // 11,010,048
    float* pyrT  = ws + 78118912ull;        // 11,010,048
    float* warps = ws + 89128960ull;        //  4,718,592
    float* comb  = ws + 93847552ull;        //  9,437,184

    float* out = (float*)d_out;             // corr_feat | corr_feat_T | imgt_pred
    float* imgt = out + 5308416;

    corr_wmma_kernel<<<dim3(64, 64, 2), 256, 0, stream>>>(fmap0, fmap1, corr, corrT);
    pyramid_kernel<<<16384, 256, 0, stream>>>(corr, corrT, pyr, pyrT);
    corr_lookup_kernel<<<(5308416 + 255) / 256, 256, 0, stream>>>(
        corr, corrT, pyr, pyrT, flow0_lr, flow1_lr, out);
    warp_combine_kernel<<<1572864 / 256, 256, 0, stream>>>(
        img0, img1, flow0, flow1, mask, img_res, mean, warps);
    conv1_kernel<<<dim3(32, 32, 2), dim3(16, 16), 0, stream>>>(warps, w1, b1, a1, comb);
    conv2_kernel<<<dim3(32, 32, 2), dim3(16, 16), 0, stream>>>(comb, w2, b2, warps, imgt);
}